// BCCEncoding_48687749267803
// MI455X (gfx1250) — compile-verified
//
#include <hip/hip_runtime.h>

// ---------------------------------------------------------------------------
// BCC tetrahedral interpolation gather kernel for MI455X (gfx1250, wave32).
// Memory-bound: ~0.4-0.8 GB of traffic, ~0.13 GFLOP -> HBM-roofline bound
// (~15-35 us at 23.3 TB/s). WMMA is mathematically inapplicable (per-point
// 1x4 * 4x8 contraction with point-varying operands). CDNA5 features used:
//   - GLOBAL_LOAD_ASYNC_TO_LDS_B128 + s_wait_asynccnt double-buffered gather
//     pipeline (gather data lands in LDS, not VGPR landing registers)
//   - NT temporal hints on the streaming pts/out traffic so the 192MB L2
//     stays dedicated to the 512MB grid (the only data with reuse)
//   - b128 vector path for all gather/store traffic, wave32-sized blocks
// ---------------------------------------------------------------------------

#define R_GRID 256
#define D_FEAT 8
#define TPB 256

typedef float v4f __attribute__((ext_vector_type(4)));
typedef int v4i __attribute__((ext_vector_type(4)));

#if defined(__has_builtin)
#if __has_builtin(__builtin_amdgcn_global_load_async_to_lds_b128) && \
    __has_builtin(__builtin_amdgcn_s_wait_asynccnt)
#define USE_ASYNC_GATHER 1
#endif
#endif

struct Geom {
  int idx[4];              // flattened grid row indices (in rows of 8 floats)
  float w1, w2, w3, w4;    // blend weights for d1..d4
};

__device__ __forceinline__ int row_index(float vx, float vy, float vz) {
  // q = clamp(floor(v * [0.5, 0.5, 1.0]), 0, R-1); idx = qx*R*R + qy*R + qz
  int qx = (int)floorf(vx * 0.5f);
  int qy = (int)floorf(vy * 0.5f);
  int qz = (int)floorf(vz);
  qx = qx < 0 ? 0 : (qx > R_GRID - 1 ? R_GRID - 1 : qx);
  qy = qy < 0 ? 0 : (qy > R_GRID - 1 ? R_GRID - 1 : qy);
  qz = qz < 0 ? 0 : (qz > R_GRID - 1 ? R_GRID - 1 : qz);
  return (qx << 16) | (qy << 8) | qz;
}

__device__ __forceinline__ Geom make_geom(float px, float py, float pz) {
  const float SXY = 2.0f * R_GRID - 1.0f;  // 511
  const float SZ = (float)(R_GRID - 1);    // 255
  float x = px * SXY, y = py * SXY, z = pz * SZ;

  float a = 0.5f * (x + y);
  float b = 0.5f * (x + z);
  float c = 0.5f * (y + z);
  float fa = floorf(a), fb = floorf(b), fc = floorf(c);
  float ra = a - fa, rb = b - fb, rc = c - fc;

  // p1 = abc_to_xyz(floors)
  float p1x = fa + fb - fc;
  float p1y = fa - fb + fc;
  float p1z = -fa + fb + fc;

  // first-occurrence argmax / argmin over (ra, rb, rc), matching jnp semantics
  int imax = (ra >= rb && ra >= rc) ? 0 : ((rb >= rc) ? 1 : 2);
  int imin = (ra <= rb && ra <= rc) ? 0 : ((rb <= rc) ? 1 : 2);

  // abc_to_xyz(onehot(k)): k=0 -> (1,1,-1); k=1 -> (1,-1,1); k=2 -> (-1,1,1)
  float e1x = (imax == 2) ? -1.0f : 1.0f;
  float e1y = (imax == 1) ? -1.0f : 1.0f;
  float e1z = (imax == 0) ? -1.0f : 1.0f;
  // abc_to_xyz(1 - onehot(k)) = (1,1,1) - abc_to_xyz(onehot(k))
  float e2x = (imin == 2) ? 2.0f : 0.0f;
  float e2y = (imin == 1) ? 2.0f : 0.0f;
  float e2z = (imin == 0) ? 2.0f : 0.0f;

  Geom g;
  g.idx[0] = row_index(p1x, p1y, p1z);
  g.idx[1] = row_index(p1x + 1.0f, p1y + 1.0f, p1z + 1.0f);
  g.idx[2] = row_index(p1x + e1x, p1y + e1y, p1z + e1z);
  g.idx[3] = row_index(p1x + e2x, p1y + e2y, p1z + e2z);

  float s1 = fmaxf(ra, fmaxf(rb, rc));
  float s3 = fminf(ra, fminf(rb, rc));
  float s2 = (ra + rb + rc) - s1 - s3;
  // out = d1 + s1*(d3-d1) + s3*(d2-d4) + s2*(d4-d3)
  //     = (1-s1)*d1 + s3*d2 + (s1-s2)*d3 + (s2-s3)*d4
  g.w1 = 1.0f - s1;
  g.w2 = s3;
  g.w3 = s1 - s2;
  g.w4 = s2 - s3;
  return g;
}

__device__ __forceinline__ v4f blend(v4f d1, v4f d2, v4f d3, v4f d4,
                                     const Geom& g) {
  v4f r = d1 * g.w1;
  r += d2 * g.w2;
  r += d3 * g.w3;
  r += d4 * g.w4;
  return r;
}

#ifdef USE_ASYNC_GATHER
// ---------------------------------------------------------------------------
// Async gather kernel (gfx1250 GLOBAL_LOAD_ASYNC_TO_LDS_B128 + ASYNCcnt):
// double-buffered LDS pipeline, 4 points/thread. All pts loads + geometry are
// hoisted to the prologue, so the steady-state loop is pure gather traffic:
// issue 8 asyncs for tile j+1 -> s_wait_asynccnt 8 -> 8 ds_load_b128 ->
// blend -> NT store.
// ---------------------------------------------------------------------------
#define PT_A 4

__device__ __forceinline__ void issue_gather(const float* __restrict__ grid,
                                             const Geom& g, v4f* dst) {
#pragma unroll
  for (int v = 0; v < 4; ++v) {
    const float* row = grid + (size_t)g.idx[v] * D_FEAT;
    __builtin_amdgcn_global_load_async_to_lds_b128(
        (v4i*)(row), (v4i*)(dst + 2 * v), 0, 0);
    __builtin_amdgcn_global_load_async_to_lds_b128(
        (v4i*)(row + 4), (v4i*)(dst + 2 * v + 1), 0, 0);
  }
}

__global__ __launch_bounds__(TPB) void bcc_async(
    const float* __restrict__ pts, const float* __restrict__ grid,
    float* __restrict__ out, long long n) {
  __shared__ v4f smem[2][TPB * 8];  // 64 KB: 2 buffers x 256 thr x 4 rows x 32B
  const int tid = threadIdx.x;
  const long long i0 = (long long)blockIdx.x * (TPB * PT_A) + tid;

  // Prologue: issue all 12 NT pts loads at once, then compute all geometry.
  float P[PT_A][3];
#pragma unroll
  for (int j = 0; j < PT_A; ++j) {
    long long i = i0 + (long long)j * TPB;
    long long ic = i < n ? i : n - 1;
    P[j][0] = __builtin_nontemporal_load(pts + 3 * ic + 0);
    P[j][1] = __builtin_nontemporal_load(pts + 3 * ic + 1);
    P[j][2] = __builtin_nontemporal_load(pts + 3 * ic + 2);
  }
  Geom g[PT_A];
#pragma unroll
  for (int j = 0; j < PT_A; ++j) g[j] = make_geom(P[j][0], P[j][1], P[j][2]);

  issue_gather(grid, g[0], &smem[0][tid * 8]);  // prime the pipeline

#pragma unroll
  for (int j = 0; j < PT_A; ++j) {
    const int buf = j & 1;
    if (j + 1 < PT_A) {
      issue_gather(grid, g[j + 1], &smem[buf ^ 1][tid * 8]);
      __builtin_amdgcn_s_wait_asynccnt(8);  // tile j landed; j+1 in flight
    } else {
      __builtin_amdgcn_s_wait_asynccnt(0);
    }
    // Each thread reads only its own LDS slots -> no barrier needed.
    const v4f* s = &smem[buf][tid * 8];
    v4f lo = blend(s[0], s[2], s[4], s[6], g[j]);
    v4f hi = blend(s[1], s[3], s[5], s[7], g[j]);
    long long i = i0 + (long long)j * TPB;
    if (i < n) {
      v4f* op = reinterpret_cast<v4f*>(out + i * D_FEAT);
      __builtin_nontemporal_store(lo, op);   // write-once stream: NT
      __builtin_nontemporal_store(hi, op + 1);
    }
  }
}

#else  // !USE_ASYNC_GATHER
// ---------------------------------------------------------------------------
// Direct-load fallback kernel: 2 points/thread, all 16 b128 loads in flight.
// ---------------------------------------------------------------------------
#define PT_D 2

__global__ __launch_bounds__(TPB) void bcc_direct(
    const float* __restrict__ pts, const float* __restrict__ grid,
    float* __restrict__ out, long long n) {
  const long long i0 = (long long)blockIdx.x * (TPB * PT_D) + threadIdx.x;

  Geom g[PT_D];
#pragma unroll
  for (int j = 0; j < PT_D; ++j) {
    long long i = i0 + (long long)j * TPB;
    long long ic = i < n ? i : n - 1;
    float px = __builtin_nontemporal_load(pts + 3 * ic + 0);
    float py = __builtin_nontemporal_load(pts + 3 * ic + 1);
    float pz = __builtin_nontemporal_load(pts + 3 * ic + 2);
    g[j] = make_geom(px, py, pz);
  }

  // Issue all gathers before any use: 16 global_load_b128 in flight / thread.
  v4f d[PT_D][4][2];
#pragma unroll
  for (int j = 0; j < PT_D; ++j) {
#pragma unroll
    for (int v = 0; v < 4; ++v) {
      const v4f* rp =
          reinterpret_cast<const v4f*>(grid + (size_t)g[j].idx[v] * D_FEAT);
      d[j][v][0] = rp[0];
      d[j][v][1] = rp[1];
    }
  }

#pragma unroll
  for (int j = 0; j < PT_D; ++j) {
    long long i = i0 + (long long)j * TPB;
    if (i < n) {
      v4f lo = blend(d[j][0][0], d[j][1][0], d[j][2][0], d[j][3][0], g[j]);
      v4f hi = blend(d[j][0][1], d[j][1][1], d[j][2][1], d[j][3][1], g[j]);
      v4f* op = reinterpret_cast<v4f*>(out + i * D_FEAT);
      __builtin_nontemporal_store(lo, op);
      __builtin_nontemporal_store(hi, op + 1);
    }
  }
}
#endif  // USE_ASYNC_GATHER

extern "C" void kernel_launch(void* const* d_in, const int* in_sizes, int n_in,
                              void* d_out, int out_size, void* d_ws,
                              size_t ws_size, hipStream_t stream) {
  const float* pts = (const float*)d_in[0];   // N x 3 f32
  const float* grid = (const float*)d_in[1];  // R^3 x 8 f32
  float* out = (float*)d_out;                 // N x 8 f32
  const long long n = (long long)in_sizes[0] / 3;

#ifdef USE_ASYNC_GATHER
  const long long per_block = (long long)TPB * PT_A;
  const int blocks = (int)((n + per_block - 1) / per_block);
  bcc_async<<<blocks, TPB, 0, stream>>>(pts, grid, out, n);
#else
  const long long per_block = (long long)TPB * PT_D;
  const int blocks = (int)((n + per_block - 1) / per_block);
  bcc_direct<<<blocks, TPB, 0, stream>>>(pts, grid, out, n);
#endif
}